// ApplyPolicyMap_5763846111936
// MI455X (gfx1250) — compile-verified
//
#include <hip/hip_runtime.h>
#include <stdint.h>

// ApplyPolicyMap for lc0 on MI455X (gfx1250).
//
// out[b, j] = x_flat[b, idx[j]] where idx[j] is the single nonzero row of
// fc1[:, j].  Pure data movement: ~474 MiB total traffic -> ~20us at 23.3 TB/s.
// Stage x rows into LDS via CDNA5 async global->LDS loads (ASYNCcnt), gather
// from LDS, store coalesced.

#define FLAT    5120     // 80 planes * 64 squares
#define NMOVES  1858
#define BLOCK   256      // 8 wave32s
#define ROWS    2        // batch rows per workgroup (2*20KB + 7.3KB LDS < 64KB)

// ---------------------------------------------------------------------------
// Kernel 1: extract the permutation from the one-hot selection matrix.
// fc1 is [FLAT, NMOVES] row-major; element e = i*NMOVES + j. Exactly one
// nonzero per column, so a plain (non-atomic) scatter write is race-free.
// ---------------------------------------------------------------------------
__global__ __launch_bounds__(256)
void build_policy_index(const float* __restrict__ fc1,
                        int* __restrict__ idx,
                        int total) {
  int e = blockIdx.x * blockDim.x + threadIdx.x;
  int stride = gridDim.x * blockDim.x;
  for (; e < total; e += stride) {
    if (fc1[e] != 0.0f) {
      idx[e % NMOVES] = e / NMOVES;
    }
  }
}

// ---------------------------------------------------------------------------
// Kernel 2: gather. Each block handles ROWS batch rows:
//   - stage ROWS * FLAT floats of x into LDS with global_load_async_to_lds_b128
//   - each wave waits its own ASYNCcnt, then workgroup barrier
//   - out[row, j] = lds_row[idx[j]]  (coalesced stores)
// ---------------------------------------------------------------------------
__global__ __launch_bounds__(BLOCK)
void policy_gather(const float* __restrict__ x,
                   const int* __restrict__ idx,
                   float* __restrict__ out) {
  __shared__ __align__(16) float sx[ROWS * FLAT];   // 40960 B
  __shared__ int sidx[NMOVES];                      //  7432 B

  const int tid = threadIdx.x;
  const long long row0 = (long long)blockIdx.x * ROWS;

  // Stage the index vector (hot in L2 across all 8192 blocks).
  for (int j = tid; j < NMOVES; j += BLOCK) {
    sidx[j] = idx[j];
  }

  // Async-copy ROWS contiguous x rows (ROWS*FLAT*4 bytes) into LDS.
  // GVS addressing: saddr = row block base (uniform), vaddr = per-lane byte
  // offset, vdst = per-lane LDS byte address. 16 B per lane per issue.
  const unsigned lds_base = (unsigned)(uintptr_t)(&sx[0]);
  const unsigned long long gbase =
      (unsigned long long)(uintptr_t)x + (unsigned long long)row0 * (FLAT * 4ull);

  unsigned off = (unsigned)tid * 16u;
#pragma unroll
  for (int it = 0; it < (ROWS * FLAT * 4) / (BLOCK * 16); ++it) {   // 10 issues
    unsigned lds_addr = lds_base + off;
    asm volatile("global_load_async_to_lds_b128 %0, %1, %2"
                 :
                 : "v"(lds_addr), "v"(off), "s"(gbase)
                 : "memory");
    off += BLOCK * 16u;
  }

  // Drain this wave's ASYNCcnt (its LDS writes are then complete), then make
  // all waves' staged data visible workgroup-wide.
  asm volatile("s_wait_asynccnt 0x0" ::: "memory");
  __syncthreads();

  // Gather + coalesced store.
#pragma unroll
  for (int r = 0; r < ROWS; ++r) {
    float* __restrict__ dst = out + (row0 + r) * NMOVES;
    const float* srow = &sx[r * FLAT];
    for (int j = tid; j < NMOVES; j += BLOCK) {
      dst[j] = srow[sidx[j]];
    }
  }
}

// ---------------------------------------------------------------------------
// Launch
// ---------------------------------------------------------------------------
extern "C" void kernel_launch(void* const* d_in, const int* in_sizes, int n_in,
                              void* d_out, int out_size, void* d_ws, size_t ws_size,
                              hipStream_t stream) {
  const float* x   = (const float*)d_in[0];   // [B, 80, 8, 8] fp32
  const float* fc1 = (const float*)d_in[1];   // [5120, 1858] fp32 one-hot cols
  float* out = (float*)d_out;                 // [B, 1858] fp32
  int* idx = (int*)d_ws;                      // NMOVES ints of scratch

  const int B = in_sizes[0] / FLAT;           // 16384
  const int total = in_sizes[1];              // 5120 * 1858

  // 1) recover the permutation from fc1 (runs every call; stateless).
  {
    int blocks = (total + 255) / 256;
    hipLaunchKernelGGL(build_policy_index, dim3(blocks), dim3(256), 0, stream,
                       fc1, idx, total);
  }

  // 2) async-staged LDS gather.
  {
    int blocks = B / ROWS;                    // 8192
    hipLaunchKernelGGL(policy_gather, dim3(blocks), dim3(BLOCK), 0, stream,
                       x, idx, out);
  }
}